// EnhancedGATRecommender_62380105007203
// MI455X (gfx1250) — compile-verified
//
#include <hip/hip_runtime.h>
#include <cstddef>

typedef __attribute__((ext_vector_type(2))) float v2f;
typedef __attribute__((ext_vector_type(8))) float v8f;

// ---------- helpers ----------
__device__ __forceinline__ int f2ord(float f) {            // order-preserving float->int
  int i = __float_as_int(f);
  return i >= 0 ? i : (i ^ 0x7fffffff);
}
__device__ __forceinline__ float ord2f(int i) {
  return __int_as_float(i >= 0 ? i : (i ^ 0x7fffffff));
}
// Non-returning device-scope f32 atomic add: force the native L2 atomic path
// (avoids any chance of a CAS-loop expansion for atomicrmw fadd).
__device__ __forceinline__ void atomAddF(float* p, float v) {
  asm volatile("global_atomic_add_f32 %0, %1, off scope:SCOPE_DEV"
               :
               : "v"(p), "v"(v)
               : "memory");
}

// ---------- GEMM: Y[n,64] = gather(X,idx) @ W^T (+bias); att[n,4] = head-dot with avec ----------
// Block = 128 threads = 4 waves; wave w computes output cols [16w,16w+16) == head w.
// Grid.x = ceil(n/16): block b computes rows [16b, 16b+16).
__global__ __launch_bounds__(128) void proj64_wmma(
    const float* __restrict__ X, const int* __restrict__ idx,
    const float* __restrict__ W, const float* __restrict__ bias,
    const float* __restrict__ avec, float* __restrict__ Y,
    float* __restrict__ att, int n)
{
  const int wave = threadIdx.x >> 5;        // output col tile / head
  const int lane = threadIdx.x & 31;
  const int half = lane >> 4;               // K-half for A/B fragments
  const int lp   = lane & 15;
  const int rb   = blockIdx.x * 16;

  int row = rb + lp;
  row = row < n ? row : n - 1;              // clamp (stores guarded later)
  const int g = idx ? idx[row] : row;
  const float* Xr = X + (size_t)g * 64;
  const float* Wr = W + (size_t)(wave * 16 + lp) * 64;   // B[k][o] = W[o][k]

  v8f acc = {0.f, 0.f, 0.f, 0.f, 0.f, 0.f, 0.f, 0.f};
#pragma unroll
  for (int k0 = 0; k0 < 64; k0 += 4) {
    // A 16x4 f32 fragment: lane(half,lp) holds A[lp][k0+2*half+{0,1}]
    v2f a = *(const v2f*)(Xr + k0 + 2 * half);
    // B 4x16 f32 fragment: lane(half,lp) holds B[k0+2*half+{0,1}][lp]
    v2f b = *(const v2f*)(Wr + k0 + 2 * half);
    acc = __builtin_amdgcn_wmma_f32_16x16x4_f32(false, a, false, b, (short)0, acc,
                                                false, false);
  }

  if (Y) {
    const float bv = bias ? bias[wave * 16 + lp] : 0.f;
#pragma unroll
    for (int v = 0; v < 8; ++v) {           // C/D: vgpr v = row 8*half+v, col lp
      const int rr = rb + half * 8 + v;
      if (rr < n) Y[(size_t)rr * 64 + wave * 16 + lp] = acc[v] + bv;
    }
  }
  if (att) {                                 // att[n][head] = sum_j P[n,16*head+j]*avec[head][j]
    const float av = avec[wave * 16 + lp];
#pragma unroll
    for (int v = 0; v < 8; ++v) {
      float t = acc[v] * av;
      t += __shfl_xor(t, 1, 32);
      t += __shfl_xor(t, 2, 32);
      t += __shfl_xor(t, 4, 32);
      t += __shfl_xor(t, 8, 32);            // reduce over 16 lanes within each half
      const int rr = rb + half * 8 + v;
      if (lp == 0 && rr < n) att[(size_t)rr * 4 + wave] = t;
    }
  }
}

// ---------- fills ----------
__global__ void fill_f32(float* __restrict__ p, float v, int n) {
  int t = blockIdx.x * blockDim.x + threadIdx.x;
  if (t < n) p[t] = v;
}
__global__ void fill_i32(int* __restrict__ p, int v, int n) {
  int t = blockIdx.x * blockDim.x + threadIdx.x;
  if (t < n) p[t] = v;
}

// ---------- edge pass 1: leaky-relu scores + segment max ----------
__global__ void edge_scores_kernel(const int* __restrict__ src, const int* __restrict__ dst,
                                   const float* __restrict__ attS, const float* __restrict__ attD,
                                   float* __restrict__ aE, int* __restrict__ amax, int E)
{
  int e = blockIdx.x * blockDim.x + threadIdx.x;
  if (e >= E) return;
  int s = src[e], d = dst[e];
#pragma unroll
  for (int h = 0; h < 4; ++h) {
    float a = attS[(size_t)s * 4 + h] + attD[(size_t)d * 4 + h];
    a = a > 0.f ? a : 0.2f * a;             // leaky_relu(., 0.2)
    aE[(size_t)e * 4 + h] = a;
    atomicMax(&amax[(size_t)d * 4 + h], f2ord(a));
  }
}

// ---------- edge pass 2: exp(a - max) + segment sum ----------
__global__ void edge_exp_kernel(const int* __restrict__ dst, float* __restrict__ aE,
                                const int* __restrict__ amax, float* __restrict__ den, int E)
{
  int e = blockIdx.x * blockDim.x + threadIdx.x;
  if (e >= E) return;
  int d = dst[e];
#pragma unroll
  for (int h = 0; h < 4; ++h) {
    float ex = expf(aE[(size_t)e * 4 + h] - ord2f(amax[(size_t)d * 4 + h]));
    aE[(size_t)e * 4 + h] = ex;             // overwrite with exp value
    atomAddF(&den[(size_t)d * 4 + h], ex);
  }
}

// ---------- edge pass 3: alpha-weighted scatter of projected src rows ----------
// 16 threads per edge; thread q handles features [4q, 4q+4) (head = q/4).
__global__ void edge_scatter_kernel(const int* __restrict__ src, const int* __restrict__ dst,
                                    const float* __restrict__ aE, const float* __restrict__ den,
                                    const float* __restrict__ P, float* __restrict__ agg, int E)
{
  int t = blockIdx.x * blockDim.x + threadIdx.x;
  int e = t >> 4;
  if (e >= E) return;
  int q = t & 15;
  int h = q >> 2;
  int s = src[e], d = dst[e];
  float alpha = aE[(size_t)e * 4 + h] / (den[(size_t)d * 4 + h] + 1e-16f);
  const float4 hv = *(const float4*)(P + (size_t)s * 64 + q * 4);
  float* ap = agg + (size_t)d * 64 + q * 4;
  atomAddF(ap + 0, hv.x * alpha);
  atomAddF(ap + 1, hv.y * alpha);
  atomAddF(ap + 2, hv.z * alpha);
  atomAddF(ap + 3, hv.w * alpha);
}

// ---------- per-relation bias broadcast ----------
__global__ void add_bias_kernel(float* __restrict__ agg, const float* __restrict__ b, int n64) {
  int t = blockIdx.x * blockDim.x + threadIdx.x;
  if (t < n64) agg[t] += b[t & 63];
}

// ---------- relu (+residual) + layernorm, row per thread ----------
__global__ void post_ln_kernel(const float* __restrict__ agg, const float* __restrict__ prev,
                               const float* __restrict__ w, const float* __restrict__ b,
                               float* __restrict__ out, int n)
{
  int r = blockIdx.x * blockDim.x + threadIdx.x;
  if (r >= n) return;
  float x[64];
  const float4* ap = (const float4*)(agg + (size_t)r * 64);
  const float4* pp = prev ? (const float4*)(prev + (size_t)r * 64) : nullptr;
  float mu = 0.f;
#pragma unroll
  for (int i = 0; i < 16; ++i) {
    float4 v = ap[i];
    float4 q;
    q.x = v.x > 0.f ? v.x : 0.f;
    q.y = v.y > 0.f ? v.y : 0.f;
    q.z = v.z > 0.f ? v.z : 0.f;
    q.w = v.w > 0.f ? v.w : 0.f;
    if (pp) { float4 pr = pp[i]; q.x += pr.x; q.y += pr.y; q.z += pr.z; q.w += pr.w; }
    x[4 * i + 0] = q.x; x[4 * i + 1] = q.y; x[4 * i + 2] = q.z; x[4 * i + 3] = q.w;
    mu += q.x + q.y + q.z + q.w;
  }
  mu *= (1.f / 64.f);
  float var = 0.f;
#pragma unroll
  for (int i = 0; i < 64; ++i) { float dd = x[i] - mu; var += dd * dd; }
  var *= (1.f / 64.f);
  float rs = rsqrtf(var + 1e-5f);
#pragma unroll
  for (int i = 0; i < 64; ++i)
    out[(size_t)r * 64 + i] = (x[i] - mu) * rs * w[i] + b[i];
}

// ======================= host side =======================

// node types: 0=user 1=song 2=artist 3=genre
static const int  kN[4]    = {100000, 200000, 20000, 500};
static const long kOff[4]  = {0, 100000, 300000, 320000};   // row offset in node buffers
// d_in leaf indices (jax pytree: dict keys sorted alphabetically)
static const int kXIdx[4]  = {0, 1, 2, 3};
static const int kEmb[4]   = {12, 11, 9, 10};               // emb: artist,genre,song,user -> 9..12
static const int kInW[4]   = {19, 17, 13, 15};
static const int kInB[4]   = {20, 18, 14, 16};
static const int kOutW[4]  = {111, 109, 105, 107};
static const int kOutB[4]  = {112, 110, 106, 108};
static const int kLn[2][2] = {{101, 102}, {103, 104}};

// per-relation layer leaves: sorted rel names: by,has,listens,performs,prefers,rev_by,rev_has,
// rev_listens,rev_performs,rev_prefers ; each {W, a_d, a_s, b}
static inline int layer_leaf(int L, int pos, int which) { return 21 + L * 40 + pos * 4 + which; }

struct RelDesc { int ei, st, dt, rev, pos; };
// reference processing order (RELS)
static const RelDesc kRels[10] = {
  {4, 0, 1, 0, 2},  // listens       user->song
  {4, 1, 0, 1, 7},  // rev_listens   song->user
  {5, 1, 2, 0, 0},  // by            song->artist
  {5, 2, 1, 1, 5},  // rev_by        artist->song
  {6, 0, 3, 0, 4},  // prefers       user->genre
  {6, 3, 0, 1, 9},  // rev_prefers   genre->user
  {7, 1, 3, 0, 1},  // has           song->genre
  {7, 3, 1, 1, 6},  // rev_has       genre->song
  {8, 2, 3, 0, 3},  // performs      artist->genre
  {8, 3, 2, 1, 8},  // rev_performs  genre->artist
};

static inline int cdiv(long a, long b) { return (int)((a + b - 1) / b); }

extern "C" void kernel_launch(void* const* d_in, const int* in_sizes, int n_in,
                              void* d_out, int out_size, void* d_ws, size_t ws_size,
                              hipStream_t stream) {
  (void)n_in; (void)out_size; (void)ws_size;
  auto F = [&](int i) { return (const float*)d_in[i]; };
  auto I = [&](int i) { return (const int*)d_in[i]; };

  const long TOT = 320500;
  float* ws = (float*)d_ws;
  size_t o = 0;
  float* bufA = ws + o; o += (size_t)TOT * 64;        // node features (ping)
  float* bufB = ws + o; o += (size_t)TOT * 64;        // node features (pong) / agg
  float* P    = ws + o; o += (size_t)200000 * 64;     // per-relation projected src rows
  float* attS = ws + o; o += (size_t)200000 * 4;
  float* attD = ws + o; o += (size_t)200000 * 4;
  int*   amax = (int*)(ws + o); o += (size_t)200000 * 4;
  float* den  = ws + o; o += (size_t)200000 * 4;
  float* aE   = ws + o; o += (size_t)800000 * 4;      // per-edge scores -> exp values

  // ---- input projection: h0[t] = emb[t][x_t] @ Win^T + bin -> bufA ----
  for (int t = 0; t < 4; ++t) {
    proj64_wmma<<<cdiv(kN[t], 16), 128, 0, stream>>>(
        F(kEmb[t]), I(kXIdx[t]), F(kInW[t]), F(kInB[t]),
        nullptr, bufA + kOff[t] * 64, nullptr, kN[t]);
  }

  float* h = bufA;
  for (int L = 0; L < 2; ++L) {
    float* agg = (L == 0) ? bufB : bufA;
    fill_f32<<<cdiv(TOT * 64, 256), 256, 0, stream>>>(agg, 0.f, (int)(TOT * 64));

    for (int r = 0; r < 10; ++r) {
      const RelDesc& R = kRels[r];
      const int E = in_sizes[R.ei] / 2;
      const int* e0  = I(R.ei);
      const int* src = R.rev ? e0 + E : e0;
      const int* dst = R.rev ? e0 : e0 + E;
      const float* W   = F(layer_leaf(L, R.pos, 0));
      const float* a_d = F(layer_leaf(L, R.pos, 1));
      const float* a_s = F(layer_leaf(L, R.pos, 2));
      const float* bb  = F(layer_leaf(L, R.pos, 3));
      const int Ns = kN[R.st], Nd = kN[R.dt];

      // src projection (stored) + per-node src attention term
      proj64_wmma<<<cdiv(Ns, 16), 128, 0, stream>>>(
          h + kOff[R.st] * 64, nullptr, W, nullptr, a_s, P, attS, Ns);
      // dst attention term only (projection discarded)
      proj64_wmma<<<cdiv(Nd, 16), 128, 0, stream>>>(
          h + kOff[R.dt] * 64, nullptr, W, nullptr, a_d, nullptr, attD, Nd);

      fill_i32<<<cdiv((long)Nd * 4, 256), 256, 0, stream>>>(amax, (int)0x80000000, Nd * 4);
      fill_f32<<<cdiv((long)Nd * 4, 256), 256, 0, stream>>>(den, 0.f, Nd * 4);

      edge_scores_kernel<<<cdiv(E, 256), 256, 0, stream>>>(src, dst, attS, attD, aE, amax, E);
      edge_exp_kernel<<<cdiv(E, 256), 256, 0, stream>>>(dst, aE, amax, den, E);
      edge_scatter_kernel<<<cdiv((long)E * 16, 256), 256, 0, stream>>>(
          src, dst, aE, den, P, agg + kOff[R.dt] * 64, E);
      add_bias_kernel<<<cdiv((long)Nd * 64, 256), 256, 0, stream>>>(
          agg + kOff[R.dt] * 64, bb, Nd * 64);
    }

    // relu (+residual on layer 2) + layernorm, in place on agg
    for (int t = 0; t < 4; ++t) {
      const float* prev = (L == 1) ? (h + kOff[t] * 64) : nullptr;
      post_ln_kernel<<<cdiv(kN[t], 256), 256, 0, stream>>>(
          agg + kOff[t] * 64, prev, F(kLn[L][0]), F(kLn[L][1]),
          agg + kOff[t] * 64, kN[t]);
    }
    h = agg;
  }

  // ---- output projection straight into d_out (user,song,artist,genre concat) ----
  float* out = (float*)d_out;
  for (int t = 0; t < 4; ++t) {
    proj64_wmma<<<cdiv(kN[t], 16), 128, 0, stream>>>(
        h + kOff[t] * 64, nullptr, F(kOutW[t]), F(kOutB[t]),
        nullptr, out + kOff[t] * 64, nullptr, kN[t]);
  }
}